// CRFLayer_9586367005401
// MI455X (gfx1250) — compile-verified
//
#include <hip/hip_runtime.h>

// CRF forward (partition function) for SEQ=1024, BATCH=1024, TAG=32 on gfx1250.
//
// Per step, per 16-batch wave:  P = expT (2x 16x32) x expA (32x16)  via
// 16 chained v_wmma_f32_16x16x4_f32 (f32 WMMA keeps the 1024-step recursion
// numerically faithful), with alpha kept register-resident in a permuted
// tag order so WMMA C/D output slots ARE the next step's B fragments.
// Log2 domain throughout (v_exp_f32 / v_log_f32 are base-2 TRANS ops).

namespace {
constexpr int   kSeq   = 1024;
constexpr int   kBatch = 1024;
constexpr int   kTag   = 32;
constexpr float kNeg   = -10000.0f;
constexpr float kLog2e = 1.4426950408889634f;
constexpr float kLn2   = 0.6931471805599453f;

typedef __attribute__((ext_vector_type(2))) float v2f;
typedef __attribute__((ext_vector_type(8))) float v8f;
} // namespace

// Swap lane L <-> L^16 within the wave (group-of-32 swizzle, xor=0x10, and=0x1f).
__device__ __forceinline__ float xor16_swap(float x) {
  int i = __builtin_amdgcn_ds_swizzle(__builtin_bit_cast(int, x), 0x401F);
  return __builtin_bit_cast(float, i);
}

__global__ __launch_bounds__(32, 1) void crf_forward_wmma(
    const float* __restrict__ feats,   // [SEQ, BATCH, TAG]
    const float* __restrict__ mask,    // [SEQ, BATCH]
    const float* __restrict__ trans,   // [TAG, TAG]  (row = out tag, col = in tag)
    float* __restrict__ out) {         // [BATCH]
  const int lane = threadIdx.x & 31;
  const int n    = lane & 15;   // batch column within the wave's 16-batch tile
  const int hL   = lane >> 4;   // lane half (WMMA layout half)
  const int b    = blockIdx.x * 16 + n;

  // Slot permutation: register slot (acc a, vgpr r, half h) holds tag
  //   sigma(a,r,h) = 4*((8a+r)>>1) + 2h + (r&1)
  // which makes the K-index mapping of the B matrix the identity.

  // --- Constant A fragments: A_a[m,k] = exp(trans[sigma(a, m&7, m>>3)][k]) ---
  // A layout (16x4 f32): lanes 0-15 -> M=lane, K = 4q+{0,1}; lanes 16-31 -> K = 4q+{2,3}.
  v2f Af[2][8];
#pragma unroll
  for (int a = 0; a < 2; ++a) {
    const int tout = 4 * ((8 * a + (n & 7)) >> 1) + 2 * (n >> 3) + (n & 1);
    const float* row = trans + tout * kTag;
#pragma unroll
    for (int q = 0; q < 8; ++q) {
      const int k0 = 4 * q + 2 * hL;
      v2f e;
      e.x = __builtin_amdgcn_exp2f(row[k0]     * kLog2e);
      e.y = __builtin_amdgcn_exp2f(row[k0 + 1] * kLog2e);
      Af[a][q] = e;
    }
  }

  // --- alpha in log2 domain, permuted-slot resident layout ---
  float alpha[2][8];
#pragma unroll
  for (int a = 0; a < 2; ++a)
#pragma unroll
    for (int r = 0; r < 8; ++r) {
      const int tag = 4 * ((8 * a + r) >> 1) + 2 * hL + (r & 1);
      alpha[a][r] = (tag == kTag - 2) ? 0.0f : kNeg * kLog2e;  // START tag = 0
    }

  const float* fbase = feats + (size_t)b * kTag;
  const float* mbase = mask + b;

  for (int t = 0; t < kSeq; ++t) {
    const float* frow = fbase + (size_t)t * (kBatch * kTag);
    if (t + 8 < kSeq)  // global_prefetch_b8 a few steps ahead of the serial scan
      __builtin_prefetch(frow + (size_t)8 * kBatch * kTag + hL * 16, 0, 0);

    // Feature pairs for this lane's 16 output slots: tags {4u+2hL, 4u+2hL+1}.
    // Streamed once -> non-temporal (do not pollute L2).
    v2f fp[8];
#pragma unroll
    for (int u = 0; u < 8; ++u)
      fp[u] = __builtin_nontemporal_load(
          reinterpret_cast<const v2f*>(frow + 4 * u + 2 * hL));
    const float mval =
        __builtin_nontemporal_load(mbase + (size_t)t * kBatch);

    // Per-batch max over 32 tags: 16 in-lane, then one cross-half swap.
    float mx = alpha[0][0];
#pragma unroll
    for (int a = 0; a < 2; ++a)
#pragma unroll
      for (int r = 0; r < 8; ++r) mx = fmaxf(mx, alpha[a][r]);
    mx = fmaxf(mx, xor16_swap(mx));

    // expA = 2^(alpha - mx)  (slots are directly the B fragments)
    float ex[2][8];
#pragma unroll
    for (int a = 0; a < 2; ++a)
#pragma unroll
      for (int r = 0; r < 8; ++r)
        ex[a][r] = __builtin_amdgcn_exp2f(alpha[a][r] - mx);

    // P = A x B : two 16x16 accumulators, K=32 in 8 steps of K=4.
    v8f p0 = {0.f, 0.f, 0.f, 0.f, 0.f, 0.f, 0.f, 0.f};
    v8f p1 = {0.f, 0.f, 0.f, 0.f, 0.f, 0.f, 0.f, 0.f};
#pragma unroll
    for (int q = 0; q < 8; ++q) {
      v2f bf;
      bf.x = ex[(2 * q) >> 3][(2 * q) & 7];          // j = 2q   -> K = 4q+2hL
      bf.y = ex[(2 * q + 1) >> 3][(2 * q + 1) & 7];  // j = 2q+1 -> K = 4q+2hL+1
      p0 = __builtin_amdgcn_wmma_f32_16x16x4_f32(false, Af[0][q], false, bf,
                                                 (short)0, p0, false, false);
      p1 = __builtin_amdgcn_wmma_f32_16x16x4_f32(false, Af[1][q], false, bf,
                                                 (short)0, p1, false, false);
    }

    // new_alpha = feat*log2e + mx + log2(P); masked blend (select, no NaN mixing).
    const bool upd = (mval != 0.0f);
#pragma unroll
    for (int a = 0; a < 2; ++a)
#pragma unroll
      for (int r = 0; r < 8; ++r) {
        const int j   = 8 * a + r;
        const float f = (r & 1) ? fp[j >> 1].y : fp[j >> 1].x;
        const float pv = (a == 0) ? p0[r] : p1[r];
        const float na =
            __builtin_fmaf(f, kLog2e, mx + __builtin_amdgcn_logf(pv));
        alpha[a][r] = upd ? na : alpha[a][r];
      }
  }

  // alpha += trans[END][tag]  (END = TAG-1), in log2 domain.
  const float* erow = trans + (kTag - 1) * kTag;
#pragma unroll
  for (int a = 0; a < 2; ++a)
#pragma unroll
    for (int r = 0; r < 8; ++r) {
      const int j = 8 * a + r;
      const float ev = erow[4 * (j >> 1) + 2 * hL + (r & 1)];
      alpha[a][r] = __builtin_fmaf(ev, kLog2e, alpha[a][r]);
    }

  // Final logsumexp over 32 tags -> natural log.
  float mx = alpha[0][0];
#pragma unroll
  for (int a = 0; a < 2; ++a)
#pragma unroll
    for (int r = 0; r < 8; ++r) mx = fmaxf(mx, alpha[a][r]);
  mx = fmaxf(mx, xor16_swap(mx));

  float s = 0.f;
#pragma unroll
  for (int a = 0; a < 2; ++a)
#pragma unroll
    for (int r = 0; r < 8; ++r) s += __builtin_amdgcn_exp2f(alpha[a][r] - mx);
  s += xor16_swap(s);

  if (hL == 0) out[b] = (mx + __builtin_amdgcn_logf(s)) * kLn2;
}

extern "C" void kernel_launch(void* const* d_in, const int* in_sizes, int n_in,
                              void* d_out, int out_size, void* d_ws,
                              size_t ws_size, hipStream_t stream) {
  (void)in_sizes; (void)n_in; (void)d_ws; (void)ws_size; (void)out_size;
  const float* feats = (const float*)d_in[0];
  const float* msk   = (const float*)d_in[1];
  const float* trans = (const float*)d_in[2];
  float* out = (float*)d_out;
  // 64 single-wave workgroups (16 batch rows per wave) -> spread across WGPs.
  crf_forward_wmma<<<kBatch / 16, 32, 0, stream>>>(feats, msk, trans, out);
}